// RBFSetConv_37847251812837
// MI455X (gfx1250) — compile-verified
//
#include <hip/hip_runtime.h>
#include <hip/hip_bf16.h>

typedef __attribute__((ext_vector_type(2))) float v2f;
typedef __attribute__((ext_vector_type(8))) float v8f;

#define GRID_RES 64
#define M_TOTAL  (GRID_RES * GRID_RES)   // 4096
#define B_SZ     8
#define N_SZ     2048
// exp(-0.5*d2/sigma^2) == exp2( (-0.5/sigma^2)*log2(e) * d2 ), sigma=0.1
#define KNEG     (-72.13475204444817f)

// Block: 256 threads = 8 waves, one batch per block; 32 blocks/batch -> 256 blocks.
// Each wave owns one 16-grid-point tile (8 waves * 32 blocks = 256 tiles = 4096 pts).
__global__ __launch_bounds__(256)
void rbf_setconv_wmma(const float* __restrict__ x_c,   // (B,N,2)
                      const float* __restrict__ y_c,   // (B,N,1)
                      const float* __restrict__ t_c,   // (B,N,1)
                      const float* __restrict__ grid,  // (1,2,M)
                      float* __restrict__ out)         // (B,3,M)
{
    // One ds_load_b128 per K-step: {x, y, yfeat, tfeat}
    __shared__ float4 sp[N_SZ];   // 32 KB

    const int block  = blockIdx.x;
    const int batch  = block >> 5;
    const int tgrp   = block & 31;
    const int tid    = threadIdx.x;
    const int wave   = tid >> 5;            // 0..7
    const int lane   = tid & 31;
    const int col    = lane & 15;           // B column / A row index
    const int half   = lane >> 4;           // K-half selector

    // ---- cooperative preload of this batch's point data into LDS ----
    {
        const float2* x2 = (const float2*)x_c + (size_t)batch * N_SZ;
        const float*  yb = y_c + (size_t)batch * N_SZ;
        const float*  tb = t_c + (size_t)batch * N_SZ;
        for (int i = tid; i < N_SZ; i += 256) {
            const float2 p = x2[i];
            sp[i] = make_float4(p.x, p.y, yb[i], tb[i]);
        }
    }
    __syncthreads();

    // ---- this wave's 16 grid points; lane handles grid column `col` ----
    const int tile = tgrp * 8 + wave;        // 0..255
    const int gm   = tile * 16 + col;
    const float gx = grid[gm];
    const float gy = grid[M_TOTAL + gm];

    v8f acc0 = {};   // rows 3..15 stay zero
    v8f acc1 = {};

    // ---- K-loop: 4 rank-4 WMMAs per trip, two independent accumulators ----
    for (int nb = 0; nb < N_SZ; nb += 16) {
#pragma unroll
        for (int u = 0; u < 4; ++u) {
            v2f avec, bvec;
#pragma unroll
            for (int j = 0; j < 2; ++j) {
                const int n = nb + 4 * u + 2 * half + j;  // K index per A/B layout
                const float4 p = sp[n];
                const float dx = p.x - gx;
                const float dy = p.y - gy;
                const float d2 = __builtin_fmaf(dy, dy, dx * dx);
                bvec[j] = __builtin_amdgcn_exp2f(KNEG * d2);          // W[k][col]
                avec[j] = (col == 0) ? 1.0f
                        : (col == 1) ? p.z
                        : (col == 2) ? p.w : 0.0f;                    // A[col][k]
            }
            if (u & 1) {
                acc1 = __builtin_amdgcn_wmma_f32_16x16x4_f32(
                    false, avec, false, bvec, (short)0, acc1, false, false);
            } else {
                acc0 = __builtin_amdgcn_wmma_f32_16x16x4_f32(
                    false, avec, false, bvec, (short)0, acc0, false, false);
            }
        }
    }

    // ---- rows 0..7 live in lanes 0..15 (VGPR r = row r); need rows 0..2 ----
    if (half == 0) {
        const float density = acc0[0] + acc1[0];
        const float wf0     = acc0[1] + acc1[1];
        const float wf1     = acc0[2] + acc1[2];
        const float dsafe   = fmaxf(density, 1e-5f);
        const float mask    = (density > 1e-5f) ? 1.0f : 0.0f;
        const float inv     = mask / dsafe;
        float* ob = out + (size_t)batch * 3 * M_TOTAL;
        ob[gm]               = density;
        ob[M_TOTAL + gm]     = wf0 * inv;
        ob[2 * M_TOTAL + gm] = wf1 * inv;
    }
}

extern "C" void kernel_launch(void* const* d_in, const int* in_sizes, int n_in,
                              void* d_out, int out_size, void* d_ws, size_t ws_size,
                              hipStream_t stream) {
    const float* x_c  = (const float*)d_in[0];   // (8,2048,2)
    const float* y_c  = (const float*)d_in[1];   // (8,2048,1)
    const float* t_c  = (const float*)d_in[2];   // (8,2048,1)
    const float* grid = (const float*)d_in[3];   // (1,2,4096)
    float* out = (float*)d_out;                  // (8,3,64,64)

    dim3 grid_dim(B_SZ * 32);   // 256 blocks
    dim3 block_dim(256);        // 8 waves
    rbf_setconv_wmma<<<grid_dim, block_dim, 0, stream>>>(x_c, y_c, t_c, grid, out);
}